// STConv_82334523064537
// MI455X (gfx1250) — compile-verified
//
#include <hip/hip_runtime.h>
#include <hip/hip_bf16.h>

// ---- problem constants (match the reference) ----
#define NB   4
#define NT   32
#define NN   1000
#define NC   64
#define NE   16000
#define NS   (NB * NT)          // 128 slices
#define ROWS (NS * NN)          // 128000 rows of [.,64]
#define EN   (NE + NN)          // edges + self loops = 17000
#define EPSV 1e-5f

typedef float v2f __attribute__((ext_vector_type(2)));
typedef float v8f __attribute__((ext_vector_type(8)));

// D = A(16x4,f32) * B(4x16,f32) + C(16x16,f32)  -- exact fp32 matrix op on CDNA5
__device__ __forceinline__ v8f wmma_f32(v2f a, v2f b, v8f c) {
    return __builtin_amdgcn_wmma_f32_16x16x4_f32(false, a, false, b, (short)0, c,
                                                 false, false);
}

// ---------------- graph normalization (tiny) ----------------
__global__ void k_deg(const int* __restrict__ ei, const float* __restrict__ ew,
                      float* __restrict__ deg) {
    int e = blockIdx.x * 256 + threadIdx.x;
    if (e >= EN) return;
    if (e < NE) atomicAdd(&deg[ei[NE + e]], ew[e]);   // col = ei[1][e]
    else        atomicAdd(&deg[e - NE], 1.0f);        // self loop, weight 1
}

__global__ void k_dis(float* __restrict__ deg) {      // in place deg -> dis
    int n = blockIdx.x * 256 + threadIdx.x;
    if (n >= NN) return;
    float d = deg[n];
    deg[n] = (d > 0.0f) ? rsqrtf(d) : 0.0f;
}

__global__ void k_norm(const int* __restrict__ ei, const float* __restrict__ ew,
                       const float* __restrict__ dis, float* __restrict__ norm) {
    int e = blockIdx.x * 256 + threadIdx.x;
    if (e >= EN) return;
    if (e < NE) norm[e] = dis[ei[e]] * ew[e] * dis[ei[NE + e]];
    else { float d = dis[e - NE]; norm[e] = d * d; }
}

// ---------------- h = X @ gcn_w  (WMMA f32 16x16x4) ----------------
// 8 waves/block, each wave: one 16-row x 64-col tile. grid = 1000.
__global__ void __launch_bounds__(256) k_gemm(const float* __restrict__ X,
                                              const float* __restrict__ W,
                                              float* __restrict__ H) {
    __shared__ float lw[NC * NC];
    for (int i = threadIdx.x; i < NC * NC; i += 256) lw[i] = W[i]; // B[k][n] layout
    __syncthreads();

    const int wave = threadIdx.x >> 5;
    const int lane = threadIdx.x & 31;
    const int m    = lane & 15;              // A row / B col within tile
    const int kp   = (lane >> 4) << 1;       // K sub-pair per half-wave
    const int mh   = (lane >> 4) << 3;       // D row offset per half-wave
    const int row0 = (blockIdx.x * 8 + wave) * 16;
    const float* arow = X + (row0 + m) * NC;

    v8f acc[4] = {};
#pragma unroll
    for (int ks = 0; ks < 16; ++ks) {
        const int k0 = ks * 4;
        v2f a = *(const v2f*)(arow + k0 + kp);
#pragma unroll
        for (int ct = 0; ct < 4; ++ct) {
            v2f b;
            b.x = lw[(k0 + kp)     * NC + ct * 16 + m];
            b.y = lw[(k0 + kp + 1) * NC + ct * 16 + m];
            acc[ct] = wmma_f32(a, b, acc[ct]);
        }
    }
#pragma unroll
    for (int ct = 0; ct < 4; ++ct)
#pragma unroll
        for (int i = 0; i < 8; ++i)
            H[(row0 + mh + i) * NC + ct * 16 + m] = acc[ct][i];
}

// ---------------- sparse scatter: agg[s,col] += norm * h[s,row] ----------------
__global__ void k_agg(const float* __restrict__ h, const int* __restrict__ ei,
                      const float* __restrict__ norm, float* __restrict__ agg) {
    int t = blockIdx.x * 256 + threadIdx.x;
    if (t >= EN * 16) return;
    const int e = t >> 4, q = (t & 15) << 2;
    const int s = blockIdx.y;
    int r, c;
    if (e < NE) { r = ei[e]; c = ei[NE + e]; } else { r = c = e - NE; }
    const float nm = norm[e];
    const float4 hv = *(const float4*)(h + (s * NN + r) * NC + q);
    float* dst = agg + (s * NN + c) * NC + q;
    atomicAdd(dst + 0, nm * hv.x);
    atomicAdd(dst + 1, nm * hv.y);
    atomicAdd(dst + 2, nm * hv.z);
    atomicAdd(dst + 3, nm * hv.w);
}

// ---------------- per-channel sum / sumsq over [ROWS,64] ----------------
__global__ void __launch_bounds__(256) k_colstats(const float* __restrict__ A,
                                                  float* __restrict__ sums,
                                                  float* __restrict__ sumsq) {
    __shared__ float ls[256], lq[256];
    const int c = threadIdx.x & 63, j = threadIdx.x >> 6;
    const int base = blockIdx.x * 256;
    float s = 0.0f, q = 0.0f;
#pragma unroll 4
    for (int i = 0; i < 64; ++i) {
        float v = A[(base + j + i * 4) * NC + c];
        s += v; q += v * v;
    }
    ls[threadIdx.x] = s; lq[threadIdx.x] = q;
    __syncthreads();
    if (j == 0) {
        float st = ls[c] + ls[64 + c] + ls[128 + c] + ls[192 + c];
        float qt = lq[c] + lq[64 + c] + lq[128 + c] + lq[192 + c];
        atomicAdd(&sums[c], st);
        atomicAdd(&sumsq[c], qt);
    }
}

// scale/shift from batch stats (bias terms cancel inside BatchNorm)
__global__ void k_finalize(const float* __restrict__ sums, const float* __restrict__ sumsq,
                           const float* __restrict__ gamma, const float* __restrict__ beta,
                           float* __restrict__ scale, float* __restrict__ shift) {
    int c = threadIdx.x;
    const float M = (float)ROWS;
    float mean = sums[c] / M;
    float var  = sumsq[c] / M - mean * mean;
    float sc   = gamma[c] * rsqrtf(var + EPSV);
    scale[c] = sc;
    shift[c] = beta[c] - mean * sc;
}

// z = relu(scale*agg + shift), in place
__global__ void __launch_bounds__(256) k_bnrelu(float* __restrict__ A,
                                                const float* __restrict__ scale,
                                                const float* __restrict__ shift) {
    int i = (blockIdx.x * 256 + threadIdx.x) * 4;
    int c = i & 63;
    float4 v = *(float4*)(A + i);
    v.x = fmaxf(scale[c + 0] * v.x + shift[c + 0], 0.0f);
    v.y = fmaxf(scale[c + 1] * v.y + shift[c + 1], 0.0f);
    v.z = fmaxf(scale[c + 2] * v.z + shift[c + 2], 0.0f);
    v.w = fmaxf(scale[c + 3] * v.w + shift[c + 3], 0.0f);
    *(float4*)(A + i) = v;
}

// ---------------- 3-tap conv along N (row-shift GEMMs, WMMA f32) ----------------
// out[row,co] = sum_kh sum_ci z[row+kh-1, ci] * w[co,ci,kh]  (zero-pad at slice edges)
__global__ void __launch_bounds__(256) k_conv(const float* __restrict__ Z,
                                              const float* __restrict__ TW,
                                              float* __restrict__ O) {
    __shared__ float lw[3 * NC * NC];            // lw[kh][ci][co] = TW[co][ci][kh]
    for (int i = threadIdx.x; i < 3 * NC * NC; i += 256) {
        int kh = i >> 12, rem = i & 4095, ci = rem >> 6, co = rem & 63;
        lw[i] = TW[(co * NC + ci) * 3 + kh];
    }
    __syncthreads();

    const int wave = threadIdx.x >> 5;
    const int lane = threadIdx.x & 31;
    const int m    = lane & 15;
    const int kp   = (lane >> 4) << 1;
    const int mh   = (lane >> 4) << 3;
    const int row0 = (blockIdx.x * 8 + wave) * 16;
    const int row  = row0 + m;
    const int nsl  = row % NN;                   // node index within slice

    v8f acc[4] = {};
#pragma unroll
    for (int kh = 0; kh < 3; ++kh) {
        const int off   = kh - 1;
        const bool valid = (unsigned)(nsl + off) < (unsigned)NN;
        const float* arow = Z + (valid ? (row + off) : row) * NC;
        const float* bw   = lw + kh * 4096;
#pragma unroll
        for (int ks = 0; ks < 16; ++ks) {
            const int k0 = ks * 4;
            v2f a = {0.0f, 0.0f};
            if (valid) a = *(const v2f*)(arow + k0 + kp);
#pragma unroll
            for (int ct = 0; ct < 4; ++ct) {
                v2f b;
                b.x = bw[(k0 + kp)     * NC + ct * 16 + m];
                b.y = bw[(k0 + kp + 1) * NC + ct * 16 + m];
                acc[ct] = wmma_f32(a, b, acc[ct]);
            }
        }
    }
#pragma unroll
    for (int ct = 0; ct < 4; ++ct)
#pragma unroll
        for (int i = 0; i < 8; ++i)
            O[(row0 + mh + i) * NC + ct * 16 + m] = acc[ct][i];
}

// out = relu(scale2*conv + shift2 + x)
__global__ void __launch_bounds__(256) k_final(const float* __restrict__ Cv,
                                               const float* __restrict__ X,
                                               const float* __restrict__ scale,
                                               const float* __restrict__ shift,
                                               float* __restrict__ out) {
    int i = (blockIdx.x * 256 + threadIdx.x) * 4;
    int c = i & 63;
    float4 v = *(const float4*)(Cv + i);
    float4 x = *(const float4*)(X + i);
    float4 o;
    o.x = fmaxf(scale[c + 0] * v.x + shift[c + 0] + x.x, 0.0f);
    o.y = fmaxf(scale[c + 1] * v.y + shift[c + 1] + x.y, 0.0f);
    o.z = fmaxf(scale[c + 2] * v.z + shift[c + 2] + x.z, 0.0f);
    o.w = fmaxf(scale[c + 3] * v.w + shift[c + 3] + x.w, 0.0f);
    *(float4*)(out + i) = o;
}

extern "C" void kernel_launch(void* const* d_in, const int* in_sizes, int n_in,
                              void* d_out, int out_size, void* d_ws, size_t ws_size,
                              hipStream_t stream) {
    const float* x     = (const float*)d_in[0];   // [B,T,N,C]
    const int*   ei    = (const int*)  d_in[1];   // [2,E]
    const float* ew    = (const float*)d_in[2];   // [E]
    const float* gcn_w = (const float*)d_in[3];   // [C,C]
    // d_in[4] = gcn_b : cancels inside BatchNorm1 (constant per channel)
    const float* g1    = (const float*)d_in[5];
    const float* b1    = (const float*)d_in[6];
    const float* tcn_w = (const float*)d_in[7];   // [C,C,3,1]
    // d_in[8] = tcn_b : cancels inside BatchNorm2
    const float* g2    = (const float*)d_in[9];
    const float* b2    = (const float*)d_in[10];
    float* out = (float*)d_out;

    float* ws   = (float*)d_ws;
    float* h    = ws;                      // 8,192,000 f  (reused as conv output)
    float* agg  = ws + (size_t)ROWS * NC;  // 8,192,000 f  (becomes z in place)
    float* deg  = agg + (size_t)ROWS * NC; // 1000 f (becomes dis)
    float* norm = deg + NN;                // 17000 f
    float* st   = norm + EN;               // stats block
    float* sum1 = st, *sq1 = st + 64, *sum2 = st + 128, *sq2 = st + 192;
    float* sc1  = st + 256, *sh1 = st + 320, *sc2 = st + 384, *sh2 = st + 448;
    float* conv = h;

    // zero: agg accumulator, deg, and the sum/sumsq block (norm in between: harmless)
    hipMemsetAsync(agg, 0, (size_t)ROWS * NC * sizeof(float), stream);
    hipMemsetAsync(deg, 0, (size_t)(NN + EN + 256) * sizeof(float), stream);

    // graph normalization
    k_deg <<<(EN + 255) / 256, 256, 0, stream>>>(ei, ew, deg);
    k_dis <<<(NN + 255) / 256, 256, 0, stream>>>(deg);
    k_norm<<<(EN + 255) / 256, 256, 0, stream>>>(ei, ew, deg, norm);

    // dense GCN GEMM (WMMA), then sparse aggregation
    k_gemm<<<ROWS / 128, 256, 0, stream>>>(x, gcn_w, h);
    dim3 gagg((EN * 16 + 255) / 256, NS);
    k_agg <<<gagg, 256, 0, stream>>>(h, ei, norm, agg);

    // BN1 + ReLU (in place on agg -> z)
    k_colstats<<<ROWS / 256, 256, 0, stream>>>(agg, sum1, sq1);
    k_finalize<<<1, 64, 0, stream>>>(sum1, sq1, g1, b1, sc1, sh1);
    k_bnrelu  <<<ROWS * NC / 1024, 256, 0, stream>>>(agg, sc1, sh1);

    // 3-tap conv along N (WMMA)
    k_conv<<<ROWS / 128, 256, 0, stream>>>(agg, tcn_w, conv);

    // BN2 + residual + ReLU
    k_colstats<<<ROWS / 256, 256, 0, stream>>>(conv, sum2, sq2);
    k_finalize<<<1, 64, 0, stream>>>(sum2, sq2, g2, b2, sc2, sh2);
    k_final   <<<ROWS * NC / 1024, 256, 0, stream>>>(conv, x, sc2, sh2, out);
}